// FlowGradientReg_77781857730942
// MI455X (gfx1250) — compile-verified
//
#include <hip/hip_runtime.h>

// Problem constants from the reference: x [B,C,H,W] f32, flow [B,H,W,2] f32.
constexpr int Bc = 32;
constexpr int Cc = 3;
constexpr int Hc = 512;
constexpr int Wc = 512;
constexpr int HWc = Hc * Wc;               // 262144 (multiple of block size)
constexpr int NPIX = Bc * HWc;             // 8388608 pixels
constexpr int LOG2_HW = 18;                // log2(512*512)
constexpr int LOG2_W = 9;                  // log2(512)
constexpr int BLOCK = 256;                 // 8 wave32 waves

// Bilinear flow-warp, one thread per (b,i,j) pixel, all C channels per thread.
// Memory-bound: ~270 MB traffic, ~12 us floor at 23.3 TB/s HBM.
// b is block-uniform (HW % BLOCK == 0) -> scalar base pointers, so the 12
// gathers + 3 stores use one 32-bit lane offset with SGPR saddr + 24-bit
// channel immediate offsets (1MB/2MB fit IOFFSET), minimizing VALU.
__global__ __launch_bounds__(BLOCK) void flow_warp_bilinear_kernel(
    const float* __restrict__ x,       // [B,C,H,W]
    const float* __restrict__ flow,    // [B,H,W,2]
    float* __restrict__ out)           // [B,C,H,W]
{
    const int base = (int)blockIdx.x * BLOCK;       // uniform
    const int b    = base >> LOG2_HW;               // uniform per block
    const int tid  = base + (int)threadIdx.x;
    const int rem  = tid & (HWc - 1);
    const int i    = rem >> LOG2_W;
    const int j    = rem & (Wc - 1);

    // Streaming prefetch of the flow input ~4KB ahead (gfx1250 global_prefetch_b8).
    __builtin_prefetch(flow + 2ull * (unsigned long long)tid + 1024, 0, 1);

    // Coalesced 8-byte flow load: flow[b,i,j,{x,y}].
    const float2 f = reinterpret_cast<const float2*>(flow)[tid];

    // Exact algebraic collapse of the reference mapping:
    // (W-1)*((-1 + 2j/(W-1)) + fx + 1)*0.5 == j + 0.5*(W-1)*fx
    const float fj = fmaf(0.5f * (float)(Wc - 1), f.x, (float)j);   // col coord
    const float fi = fmaf(0.5f * (float)(Hc - 1), f.y, (float)i);   // row coord

    // floor -> clip (reference order); weights use the *clipped* lower corner.
    const float i1f = fminf(fmaxf(floorf(fi), 0.0f), (float)(Hc - 1));
    const float i2f = fminf(i1f + 1.0f, (float)(Hc - 1));
    const float j1f = fminf(fmaxf(floorf(fj), 0.0f), (float)(Wc - 1));
    const float j2f = fminf(j1f + 1.0f, (float)(Wc - 1));

    const int i1 = (int)i1f, i2 = (int)i2f;
    const int j1 = (int)j1f, j2 = (int)j2f;

    const float di  = fi - i1f;
    const float dj  = fj - j1f;
    const float odi = 1.0f - di;
    const float odj = 1.0f - dj;

    // 32-bit element indices; channel stride folds into the 24-bit IOFFSET.
    const int o11 = (i1 << LOG2_W) + j1;
    const int o12 = (i1 << LOG2_W) + j2;
    const int o21 = (i2 << LOG2_W) + j1;
    const int o22 = (i2 << LOG2_W) + j2;

    const float* __restrict__ xb = x   + (size_t)(b * Cc) * (size_t)HWc; // SGPR base
    float* __restrict__       ob = out + (size_t)(b * Cc) * (size_t)HWc; // SGPR base

#pragma unroll
    for (int c = 0; c < Cc; ++c) {
        const int cs = c * HWc;                     // compile-time immediate
        const float q11 = xb[cs + o11];
        const float q12 = xb[cs + o12];
        const float q21 = xb[cs + o21];
        const float q22 = xb[cs + o22];
        const float qi1 = fmaf(q21, di, q11 * odi);
        const float qi2 = fmaf(q22, di, q12 * odi);
        ob[cs + rem] = fmaf(qi2, dj, qi1 * odj);    // coalesced store
    }
}

extern "C" void kernel_launch(void* const* d_in, const int* in_sizes, int n_in,
                              void* d_out, int out_size, void* d_ws, size_t ws_size,
                              hipStream_t stream) {
    (void)in_sizes; (void)n_in; (void)out_size; (void)d_ws; (void)ws_size;
    const float* x    = reinterpret_cast<const float*>(d_in[0]);  // [B,C,H,W] f32
    const float* flow = reinterpret_cast<const float*>(d_in[1]);  // [B,H,W,2] f32
    float* out        = reinterpret_cast<float*>(d_out);          // [B,C,H,W] f32

    constexpr int grid = NPIX / BLOCK;               // 32768 blocks, exact
    flow_warp_bilinear_kernel<<<grid, BLOCK, 0, stream>>>(x, flow, out);
}